// Recommender_39084202394386
// MI455X (gfx1250) — compile-verified
//
#include <hip/hip_runtime.h>
#include <math.h>

typedef float v2f __attribute__((ext_vector_type(2)));
typedef float v8f __attribute__((ext_vector_type(8)));

#define BATCH  128
#define QQ     20
#define DD     11
#define QD     220            // QQ*DD
#define NSONGS 100000
#define PK     12             // K=11 padded to 12 (3 WMMA k-steps of 4)
#define MROWS  (BATCH * QQ)   // 2560
#define MTILES (MROWS / 16)   // 160
#define NTILES (NSONGS / 16)  // 6250
#define EPSN   1e-8f

// ---------------------------------------------------------------------------
// Kernel 1: encoder MLP + L2-normalize "produced" into padded [MROWS][PK]
// One block of 128 threads; thread b handles sample b. h1 staged in LDS
// transposed (h1s[j*128+b]) so every access is bank-conflict free.
// Unrolling capped: this kernel is ~30 KFLOP/thread, launched once — keep it
// compact instead of letting the compiler emit thousands of ds/valu ops.
// ---------------------------------------------------------------------------
__global__ __launch_bounds__(128)
void mlp_norm_kernel(const float* __restrict__ x,
                     const float* __restrict__ W1, const float* __restrict__ b1,
                     const float* __restrict__ W2, const float* __restrict__ b2,
                     const float* __restrict__ W3, const float* __restrict__ b3,
                     float* __restrict__ pn)
{
    __shared__ float h1s[128 * 128];   // 64 KB of the 320 KB WGP LDS
    const int b = threadIdx.x;

    float xv[55];
#pragma unroll
    for (int i = 0; i < 55; ++i) xv[i] = x[b * 55 + i];

#pragma unroll 2
    for (int j = 0; j < 128; ++j) {
        float acc = b1[j];
#pragma unroll
        for (int i = 0; i < 55; ++i) acc = fmaf(xv[i], W1[i * 128 + j], acc);
        h1s[j * 128 + b] = fmaxf(acc, 0.0f);   // ReLU
    }

    float h2[64];
#pragma unroll 2
    for (int j = 0; j < 64; ++j) {
        float acc = b2[j];
#pragma unroll 4
        for (int i = 0; i < 128; ++i) acc = fmaf(h1s[i * 128 + b], W2[i * 64 + j], acc);
        h2[j] = fmaxf(acc, 0.0f);              // ReLU
    }

#pragma unroll 1
    for (int q = 0; q < QQ; ++q) {
        float o[DD];
        float ss = 0.0f;
#pragma unroll
        for (int d = 0; d < DD; ++d) {
            float acc = b3[q * DD + d];
#pragma unroll 4
            for (int j = 0; j < 64; ++j) acc = fmaf(h2[j], W3[j * QD + q * DD + d], acc);
            o[d] = acc;
            ss = fmaf(acc, acc, ss);
        }
        const float inv = 1.0f / fmaxf(sqrtf(ss), EPSN);
        float* dst = pn + (b * QQ + q) * PK;
#pragma unroll
        for (int d = 0; d < DD; ++d) dst[d] = o[d] * inv;
        dst[DD] = 0.0f;                        // zero pad lane K=11
    }
}

// ---------------------------------------------------------------------------
// Kernel 2: L2-normalize songs into padded [NSONGS][PK]
// ---------------------------------------------------------------------------
__global__ __launch_bounds__(256)
void norm_songs_kernel(const float* __restrict__ songs, float* __restrict__ sn)
{
    const int n = blockIdx.x * blockDim.x + threadIdx.x;
    if (n >= NSONGS) return;
    float v[DD];
    float ss = 0.0f;
#pragma unroll
    for (int d = 0; d < DD; ++d) {
        v[d] = songs[n * DD + d];
        ss = fmaf(v[d], v[d], ss);
    }
    const float inv = 1.0f / fmaxf(sqrtf(ss), EPSN);
    float* dst = sn + n * PK;
#pragma unroll
    for (int d = 0; d < DD; ++d) dst[d] = v[d] * inv;
    dst[DD] = 0.0f;
}

// ---------------------------------------------------------------------------
// Kernel 3: streaming WMMA similarity + fused argmax + gather.
// grid.x = MTILES (one 16-row M tile per block), 256 threads = 8 waves.
// Wave w owns N-tiles {w, w+8, ...}. Per tile: 3x v_wmma_f32_16x16x4_f32.
// C layout: lane l, VGPR r  <->  sim[m0 + r + 8*(l>>4)][n0 + (l&15)].
// The song fragments for tile t+8 are double-buffered in registers so the
// load-wait lands a full iteration later (software pipelining). The pipeline
// guard is wave-uniform, so EXEC is all-1s at every WMMA.
// ---------------------------------------------------------------------------
__global__ __launch_bounds__(256)
void knn_argmax_kernel(const float* __restrict__ pn,
                       const float* __restrict__ sn,
                       const float* __restrict__ songs,
                       float* __restrict__ out)
{
    __shared__ float smax[8][16];
    __shared__ int   sidx[8][16];

    const int lane = threadIdx.x & 31;
    const int wave = threadIdx.x >> 5;    // 0..7
    const int half = lane >> 4;           // 0 or 1
    const int lm   = lane & 15;
    const int m0   = blockIdx.x * 16;
    const int koff = 2 * half;            // this lane's K sub-offset

    // A fragments (constant over the N stream): lane holds row m0+lm,
    // VGPR pair = {K = 4t + 2*half, K = 4t + 2*half + 1}
    const float* prow = pn + (m0 + lm) * PK;
    v2f a0, a1, a2;
    a0.x = prow[0 + koff];  a0.y = prow[0 + koff + 1];
    a1.x = prow[4 + koff];  a1.y = prow[4 + koff + 1];
    a2.x = prow[8 + koff];  a2.y = prow[8 + koff + 1];

    float best[8];
    int   bidx[8];
#pragma unroll
    for (int r = 0; r < 8; ++r) { best[r] = -3.4e38f; bidx[r] = 0x7fffffff; }

    // Prologue: load fragments for this wave's first tile (NTILES >> 8, so
    // every wave has at least one tile).
    int tile = wave;
    v2f b0, b1, b2;
    {
        const float* srow = sn + (tile * 16 + lm) * PK;
        b0.x = srow[0 + koff];  b0.y = srow[0 + koff + 1];
        b1.x = srow[4 + koff];  b1.y = srow[4 + koff + 1];
        b2.x = srow[8 + koff];  b2.y = srow[8 + koff + 1];
    }

    while (tile < NTILES) {
        const int next = tile + 8;

        // Stage next tile's fragments (wave-uniform guard — no divergence).
        v2f c0 = b0, c1 = b1, c2 = b2;
        if (next < NTILES) {
            const float* srow = sn + (next * 16 + lm) * PK;
            __builtin_prefetch(sn + ((tile + 16) * 16 + lm) * PK, 0, 3);
            c0.x = srow[0 + koff];  c0.y = srow[0 + koff + 1];
            c1.x = srow[4 + koff];  c1.y = srow[4 + koff + 1];
            c2.x = srow[8 + koff];  c2.y = srow[8 + koff + 1];
        }

        v8f acc = {};
        acc = __builtin_amdgcn_wmma_f32_16x16x4_f32(false, a0, false, b0, (short)0, acc, false, false);
        acc = __builtin_amdgcn_wmma_f32_16x16x4_f32(false, a1, false, b1, (short)0, acc, false, false);
        acc = __builtin_amdgcn_wmma_f32_16x16x4_f32(false, a2, false, b2, (short)0, acc, false, false);

        const int n = tile * 16 + lm;
#pragma unroll
        for (int r = 0; r < 8; ++r) {
            if (acc[r] > best[r]) { best[r] = acc[r]; bidx[r] = n; }  // strict > => first index wins
        }

        b0 = c0;  b1 = c1;  b2 = c2;
        tile = next;
    }

    // Reduce over the 16 N-lanes inside each half-wave (xor 1,2,4,8 stays in half).
#pragma unroll
    for (int off = 1; off < 16; off <<= 1) {
#pragma unroll
        for (int r = 0; r < 8; ++r) {
            const float om = __shfl_xor(best[r], off, 32);
            const int   oi = __shfl_xor(bidx[r], off, 32);
            if (om > best[r] || (om == best[r] && oi < bidx[r])) { best[r] = om; bidx[r] = oi; }
        }
    }

    // Lane lm==0 of each half publishes its 8 m-rows (m = r + 8*half).
    if (lm == 0) {
#pragma unroll
        for (int r = 0; r < 8; ++r) {
            smax[wave][r + half * 8] = best[r];
            sidx[wave][r + half * 8] = bidx[r];
        }
    }
    __syncthreads();

    // Threads 0..15: combine the 8 wave-partials for row m, gather songs[best].
    if (threadIdx.x < 16) {
        const int m = threadIdx.x;
        float fm = smax[0][m];
        int   fi = sidx[0][m];
#pragma unroll
        for (int w = 1; w < 8; ++w) {
            const float om = smax[w][m];
            const int   oi = sidx[w][m];
            if (om > fm || (om == fm && oi < fi)) { fm = om; fi = oi; }
        }
        const float* src = songs + (long)fi * DD;
        float* dst = out + (long)(m0 + m) * DD;
#pragma unroll
        for (int d = 0; d < DD; ++d) dst[d] = src[d];
    }
}

// ---------------------------------------------------------------------------
// Launch wrapper
// ---------------------------------------------------------------------------
extern "C" void kernel_launch(void* const* d_in, const int* in_sizes, int n_in,
                              void* d_out, int out_size, void* d_ws, size_t ws_size,
                              hipStream_t stream)
{
    const float* x     = (const float*)d_in[0];
    const float* W1    = (const float*)d_in[1];
    const float* b1    = (const float*)d_in[2];
    const float* W2    = (const float*)d_in[3];
    const float* b2    = (const float*)d_in[4];
    const float* W3    = (const float*)d_in[5];
    const float* b3    = (const float*)d_in[6];
    const float* songs = (const float*)d_in[7];
    float* out = (float*)d_out;

    // Workspace: padded p_norm [2560][12] then padded s_norm [100000][12] (~4.9 MB)
    float* pn = (float*)d_ws;
    float* sn = pn + (size_t)MROWS * PK;

    mlp_norm_kernel<<<1, 128, 0, stream>>>(x, W1, b1, W2, b2, W3, b3, pn);
    norm_songs_kernel<<<(NSONGS + 255) / 256, 256, 0, stream>>>(songs, sn);
    knn_argmax_kernel<<<MTILES, 256, 0, stream>>>(pn, sn, songs, out);
}